// Edge_net_13795434954992
// MI455X (gfx1250) — compile-verified
//
#include <hip/hip_runtime.h>
#include <math.h>

// Reference dims
#define IN_F 64
#define OUT_F 64
#define NTYPES 10
#define MB 2                 // M-tiles (16 edges each) per wave
#define SEG_ALIGN 32         // segment padding so 32-slot wave-tiles are type-uniform

typedef float v2f __attribute__((ext_vector_type(2)));
typedef float v8f __attribute__((ext_vector_type(8)));

#if __has_builtin(__builtin_amdgcn_tanhf)
#define TANH_F32(x) __builtin_amdgcn_tanhf(x)
#else
#define TANH_F32(x) tanhf(x)
#endif

// -------- workspace layout (ints) --------
// ws[0..9]    : per-type counts
// ws[10..20]  : 32-aligned segment offsets (11 entries, last = padded total)
// ws[21..30]  : scatter cursors
// ws[32..]    : perm[E + SEG_ALIGN*NTYPES]  (edge id per sorted slot, -1 = padding)

__global__ void init_ws_kernel(int* __restrict__ ws, int nperm) {
    int i = blockIdx.x * blockDim.x + threadIdx.x;
    if (i < 32) ws[i] = 0;
    if (i < nperm) ws[32 + i] = -1;
}

__global__ void hist_kernel(const int* __restrict__ type, int* __restrict__ ws, int E) {
    int i = blockIdx.x * blockDim.x + threadIdx.x;
    if (i < E) atomicAdd(&ws[type[i]], 1);
}

__global__ void scan_kernel(int* __restrict__ ws) {
    if (threadIdx.x == 0) {
        int off = 0;
        for (int t = 0; t < NTYPES; ++t) {
            ws[10 + t] = off;          // segment start (32-aligned)
            ws[21 + t] = off;          // cursor
            off += (ws[t] + (SEG_ALIGN - 1)) & ~(SEG_ALIGN - 1);
        }
        ws[20] = off;                  // padded total
    }
}

__global__ void scatter_kernel(const int* __restrict__ type, int* __restrict__ ws,
                               int* __restrict__ perm, int E) {
    int i = blockIdx.x * blockDim.x + threadIdx.x;
    if (i < E) {
        int t = type[i];
        int pos = atomicAdd(&ws[21 + t], 1);
        perm[pos] = i;
    }
}

// One wave per 32-edge (2 x 16-row M-tile) type-uniform tile.
// Each loaded B fragment (W_t^T and lin_w^T) feeds MB WMMAs.
// Padding rows: only the *address* falls back to a valid edge (e0); the
// resulting garbage D rows are row-independent and never stored, so no
// data zeroing (cndmask) is needed.
__global__ __launch_bounds__(256) void edge_gemm_kernel(
    const float* __restrict__ desc,    // [E, 64]
    const float* __restrict__ layer1,  // [10, 64, 64]
    const float* __restrict__ lin_w,   // [64, 64]
    const float* __restrict__ lin_b,   // [64]
    const int*   __restrict__ type,    // [E]
    const int*   __restrict__ perm,    // [ntiles*32]
    float*       __restrict__ out,     // [E, 64]
    int ntiles)
{
    const int wave = threadIdx.x >> 5;
    const int lane = threadIdx.x & 31;
    const int tile = blockIdx.x * 8 + wave;
    if (tile >= ntiles) return;

    const int slot  = tile * (16 * MB);
    const int m     = lane & 15;   // A row / B col within a 16x16 tile
    const int khalf = lane >> 4;   // K sub-pair selector (0 or 1)

    const int e0 = perm[slot];     // slot 0 of any live tile is valid (segments 32-padded)
    if (e0 < 0) return;            // fully-padded tail tile (wave-uniform exit)

    const int t = type[e0];        // uniform across wave

    // ---- A fragments for both M-tiles: 16 K-steps x v2f each ----
    v2f a[MB][16];
#pragma unroll
    for (int mt = 0; mt < MB; ++mt) {
        const int em = perm[slot + mt * 16 + m];
        // Address-only fallback for padding rows; their D rows are discarded.
        const float* aptr = desc + (size_t)(em >= 0 ? em : e0) * IN_F + 2 * khalf;
#pragma unroll
        for (int s = 0; s < 16; ++s) {
            a[mt][s] = *(const v2f*)(aptr + 4 * s);
        }
    }

    // ---- output row indices (D tile: vgpr r -> M = r + 8*khalf) ----
    int eRow[MB][8];
#pragma unroll
    for (int mt = 0; mt < MB; ++mt)
#pragma unroll
        for (int r = 0; r < 8; ++r)
            eRow[mt][r] = perm[slot + mt * 16 + r + 8 * khalf];

    const float* wt = layer1 + (size_t)t * OUT_F * IN_F;

#pragma unroll
    for (int nb = 0; nb < 4; ++nb) {
        // B fragment (4x16, KxN): lane holds col n=m, K pair 2*khalf; B = W^T.
        const float* wrow = wt    + (size_t)(nb * 16 + m) * IN_F + 2 * khalf;
        const float* lrow = lin_w + (size_t)(nb * 16 + m) * IN_F + 2 * khalf;

        v8f accW[MB], accL[MB];
#pragma unroll
        for (int mt = 0; mt < MB; ++mt) {
            accW[mt] = (v8f){0.f, 0.f, 0.f, 0.f, 0.f, 0.f, 0.f, 0.f};
            accL[mt] = (v8f){0.f, 0.f, 0.f, 0.f, 0.f, 0.f, 0.f, 0.f};
        }
#pragma unroll
        for (int s = 0; s < 16; ++s) {
            const v2f bw = *(const v2f*)(wrow + 4 * s);
            const v2f bl = *(const v2f*)(lrow + 4 * s);
#pragma unroll
            for (int mt = 0; mt < MB; ++mt) {
                accW[mt] = __builtin_amdgcn_wmma_f32_16x16x4_f32(
                    false, a[mt][s], false, bw, (short)0, accW[mt], false, false);
                accL[mt] = __builtin_amdgcn_wmma_f32_16x16x4_f32(
                    false, a[mt][s], false, bl, (short)0, accL[mt], false, false);
            }
        }

        // Epilogue for this N-tile: tanh(bond GEMM) + linear adjust + bias.
        const float bias = lin_b[nb * 16 + m];
#pragma unroll
        for (int mt = 0; mt < MB; ++mt) {
            const v8f cw = accW[mt];
            const v8f cl = accL[mt];
#pragma unroll
            for (int r = 0; r < 8; ++r) {
                const int e = eRow[mt][r];
                if (e >= 0) {
                    out[(size_t)e * OUT_F + nb * 16 + m] = TANH_F32(cw[r]) + cl[r] + bias;
                }
            }
        }
    }
}

extern "C" void kernel_launch(void* const* d_in, const int* in_sizes, int n_in,
                              void* d_out, int out_size, void* d_ws, size_t ws_size,
                              hipStream_t stream) {
    const int*   batch_ij = (const int*)d_in[0];    // [E] int32
    const float* desc     = (const float*)d_in[1];  // [E,64,1] fp32
    const float* layer1   = (const float*)d_in[2];  // [10,64,64] fp32
    const float* lin_w    = (const float*)d_in[3];  // [64,64] fp32
    const float* lin_b    = (const float*)d_in[4];  // [64] fp32
    float*       out      = (float*)d_out;          // [E,64,1] fp32

    const int E     = in_sizes[0];
    const int nperm = E + SEG_ALIGN * NTYPES;       // worst-case padded slots
    int* ws   = (int*)d_ws;
    int* perm = ws + 32;

    // 1) reset header + padding sentinel
    init_ws_kernel<<<(nperm + 255) / 256, 256, 0, stream>>>(ws, nperm);
    // 2) per-type histogram
    hist_kernel<<<(E + 255) / 256, 256, 0, stream>>>(batch_ij, ws, E);
    // 3) 32-aligned exclusive scan over 10 types
    scan_kernel<<<1, 32, 0, stream>>>(ws);
    // 4) scatter edge ids into type-sorted, 32-padded order
    scatter_kernel<<<(E + 255) / 256, 256, 0, stream>>>(batch_ij, ws, perm, E);
    // 5) segmented WMMA GEMM (2 M-tiles/wave) + tanh epilogue
    const int ntiles = nperm / (16 * MB);
    edge_gemm_kernel<<<(ntiles + 7) / 8, 256, 0, stream>>>(
        desc, layer1, lin_w, lin_b, batch_ij, perm, out, ntiles);
}